// HardNegativeLoss_60344290509134
// MI455X (gfx1250) — compile-verified
//
#include <hip/hip_runtime.h>
#include <hip/hip_bf16.h>
#include <stdint.h>

// Problem constants (from the reference)
#define NROWS 4096
#define DIM   1024
#define NTILE (NROWS / 64)
#define TILE  64          // rows/cols of distance tile per block
#define DK    16          // depth chunk staged in LDS per iteration
#define NCHUNK (DIM / DK) // 64

#ifndef __has_builtin
#define __has_builtin(x) 0
#endif

// ---------------- CDNA5 async global->LDS copy (ASYNCcnt path) ----------------
#if __has_builtin(__builtin_amdgcn_global_load_async_to_lds_b32)
typedef __attribute__((address_space(1))) int GAS1I; // global int
typedef __attribute__((address_space(3))) int LAS3I; // LDS int
#define ASYNC_CP_B32(GP, LP)                                                   \
  __builtin_amdgcn_global_load_async_to_lds_b32(                               \
      (GAS1I*)(uintptr_t)(GP), (LAS3I*)(uint32_t)(uintptr_t)(LP), 0, 0)
#else
// gfx1250 asm: global_load_async_to_lds_b32 vdst(LDS addr), vaddr(64b), off
#define ASYNC_CP_B32(GP, LP)                                                   \
  asm volatile("global_load_async_to_lds_b32 %0, %1, off"                      \
               :: "v"((uint32_t)(uintptr_t)(LP)), "v"(GP) : "memory")
#endif

#if __has_builtin(__builtin_amdgcn_s_wait_asynccnt)
#define WAIT_ASYNC(N) __builtin_amdgcn_s_wait_asynccnt(N)
#else
#define WAIT_ASYNC(N) asm volatile("s_wait_asynccnt %0" :: "i"(N) : "memory")
#endif

static __device__ __forceinline__ unsigned long long umin64(unsigned long long a,
                                                            unsigned long long b) {
  return a < b ? a : b;
}

// ------------------- tiled Chebyshev + masked argmin (hot kernel) -------------------
// Placed first in the file so the disasm snippet shows its async-copy prologue.
__global__ __launch_bounds__(256)
void hn_tile_kernel(const float* __restrict__ F, const int* __restrict__ labels,
                    unsigned long long* __restrict__ packed) {
  const int bi = blockIdx.y;   // row-tile
  const int bj = blockIdx.x;   // col-tile
  if (bi > bj) return;         // symmetry: dist[i,j] == dist[j,i]

  __shared__ __align__(16) float As[2][DK][TILE]; // d-major: bank-conflict-free b128 reads
  __shared__ __align__(16) float Bs[2][DK][TILE];
  __shared__ int labI[TILE], labJ[TILE];
  __shared__ unsigned long long colI[TILE], colJ[TILE];

  const int tid = threadIdx.x;
  const int tx = tid & 15;     // j sub-tile
  const int ty = tid >> 4;     // i sub-tile
  const int i0 = bi * TILE, j0 = bj * TILE;

  if (tid < TILE) {
    labI[tid] = labels[i0 + tid];
    labJ[tid] = labels[j0 + tid];
    colI[tid] = ~0ull;
  } else if (tid < 2 * TILE) {
    colJ[tid - TILE] = ~0ull;
  }

  // per-lane async copy pattern: lane covers (r = (tid>>4)+16k, d = tid&15), k=0..3
  const int dcol  = tid & 15;
  const int rbase = tid >> 4;
#define ISSUE_CHUNK(C, B)                                                      \
  do {                                                                         \
    _Pragma("unroll")                                                          \
    for (int k = 0; k < 4; ++k) {                                              \
      const int r = rbase + k * 16;                                            \
      ASYNC_CP_B32(F + (size_t)(i0 + r) * DIM + (C) * DK + dcol,               \
                   &As[(B)][dcol][r]);                                         \
      ASYNC_CP_B32(F + (size_t)(j0 + r) * DIM + (C) * DK + dcol,               \
                   &Bs[(B)][dcol][r]);                                         \
    }                                                                          \
  } while (0)

  float acc[4][4] = {{0.f,0.f,0.f,0.f},{0.f,0.f,0.f,0.f},
                     {0.f,0.f,0.f,0.f},{0.f,0.f,0.f,0.f}};

  ISSUE_CHUNK(0, 0);
  int buf = 0;
  for (int c = 0; c < NCHUNK; ++c) {
    if (c + 1 < NCHUNK) {
      ISSUE_CHUNK(c + 1, buf ^ 1);  // prefetch next chunk (8 more asyncs in flight)
      WAIT_ASYNC(8);                // first 8 (current chunk) complete, in order
    } else {
      WAIT_ASYNC(0);
    }
    __syncthreads();

#pragma unroll
    for (int d = 0; d < DK; ++d) {
      const float4 av = *(const float4*)&As[buf][d][4 * ty];
      const float4 bv = *(const float4*)&Bs[buf][d][4 * tx];
      const float a[4] = {av.x, av.y, av.z, av.w};
      const float b[4] = {bv.x, bv.y, bv.z, bv.w};
#pragma unroll
      for (int ii = 0; ii < 4; ++ii)
#pragma unroll
        for (int jj = 0; jj < 4; ++jj)
          acc[ii][jj] = fmaxf(acc[ii][jj], fabsf(a[ii] - b[jj])); // v_sub + v_max|.|
    }
    __syncthreads(); // everyone done reading buf before it is overwritten
    buf ^= 1;
  }

  // label-masked packed argmin: key = (float_bits(dist)<<32)|index (nonneg floats
  // order like uints; ties -> smallest index == jnp.argmin first occurrence)
  int li[4], lj[4];
#pragma unroll
  for (int k = 0; k < 4; ++k) { li[k] = labI[4 * ty + k]; lj[k] = labJ[4 * tx + k]; }

#pragma unroll
  for (int jj = 0; jj < 4; ++jj) { // candidates (i) for column j0+4tx+jj
    unsigned long long best = ~0ull;
#pragma unroll
    for (int ii = 0; ii < 4; ++ii)
      if (li[ii] != lj[jj])
        best = umin64(best, ((unsigned long long)__float_as_uint(acc[ii][jj]) << 32)
                                | (unsigned)(i0 + 4 * ty + ii));
    if (best != ~0ull) atomicMin(&colJ[4 * tx + jj], best);
  }
#pragma unroll
  for (int ii = 0; ii < 4; ++ii) { // symmetric candidates (j) for column i0+4ty+ii
    unsigned long long best = ~0ull;
#pragma unroll
    for (int jj = 0; jj < 4; ++jj)
      if (li[ii] != lj[jj])
        best = umin64(best, ((unsigned long long)__float_as_uint(acc[ii][jj]) << 32)
                                | (unsigned)(j0 + 4 * tx + jj));
    if (best != ~0ull) atomicMin(&colI[4 * ty + ii], best);
  }
  __syncthreads();

  if (tid < TILE) {
    const unsigned long long v = colJ[tid];
    if (v != ~0ull) atomicMin(&packed[j0 + tid], v);
  } else if (tid < 2 * TILE) {
    const unsigned long long v = colI[tid - TILE];
    if (v != ~0ull) atomicMin(&packed[i0 + (tid - TILE)], v);
  }
#undef ISSUE_CHUNK
}

// ------------------------- per-row max normalization -------------------------
__global__ __launch_bounds__(256)
void hn_norm_kernel(const float* __restrict__ feats, const int* __restrict__ flag,
                    float* __restrict__ out) {
  __shared__ float red[256];
  const int row = blockIdx.x;
  const int tid = threadIdx.x;
  const float4 x = ((const float4*)(feats + (size_t)row * DIM))[tid]; // 256*4 == DIM
  float mx = fmaxf(fmaxf(x.x, x.y), fmaxf(x.z, x.w));
  red[tid] = mx;
  __syncthreads();
#pragma unroll
  for (int s = 128; s > 0; s >>= 1) {
    if (tid < s) red[tid] = fmaxf(red[tid], red[tid + s]);
    __syncthreads();
  }
  const float rmax = red[0];
  float4 y;
  if (flag[0] != 0) {
    y.x = x.x / rmax; y.y = x.y / rmax; y.z = x.z / rmax; y.w = x.w / rmax;
  } else {
    y = x;
  }
  ((float4*)(out + (size_t)row * DIM))[tid] = y;
}

// ------------------------- init packed (distbits<<32 | idx) -------------------------
__global__ __launch_bounds__(256)
void hn_init_kernel(unsigned long long* __restrict__ packed) {
  const int i = blockIdx.x * 256 + threadIdx.x;
  if (i < NROWS) packed[i] = ~0ull;
}

// ------------------------- finalize: loss mean + pair indices -------------------------
__global__ __launch_bounds__(256)
void hn_final_kernel(const unsigned long long* __restrict__ packed,
                     const int* __restrict__ image_idxs, float* __restrict__ out) {
  __shared__ float red[256];
  const int tid = threadIdx.x;
  float s = 0.f;
  for (int k = tid; k < NROWS; k += 256) {
    const unsigned long long v = packed[k];
    const unsigned idx = (unsigned)(v & 0xffffffffull);
    s += __uint_as_float((unsigned)(v >> 32));
    out[1 + k] = (float)image_idxs[idx]; // closest_pairs (exact in f32, idx < 4096)
  }
  red[tid] = s;
  __syncthreads();
#pragma unroll
  for (int t = 128; t > 0; t >>= 1) {
    if (tid < t) red[tid] += red[tid + t];
    __syncthreads();
  }
  if (tid == 0) out[0] = red[0] / (float)NROWS;
}

extern "C" void kernel_launch(void* const* d_in, const int* in_sizes, int n_in,
                              void* d_out, int out_size, void* d_ws, size_t ws_size,
                              hipStream_t stream) {
  const float* feats      = (const float*)d_in[0];
  const int*   labels     = (const int*)d_in[1];
  const int*   image_idxs = (const int*)d_in[2];
  const int*   normflag   = (const int*)d_in[3];
  float* out = (float*)d_out;

  char* ws = (char*)d_ws;
  float* normed = (float*)ws;                                  // 4096*1024*4 = 16 MB
  unsigned long long* packed =
      (unsigned long long*)(ws + (size_t)NROWS * DIM * sizeof(float)); // 32 KB

  hn_norm_kernel<<<NROWS, 256, 0, stream>>>(feats, normflag, normed);
  hn_init_kernel<<<(NROWS + 255) / 256, 256, 0, stream>>>(packed);
  dim3 grid(NTILE, NTILE); // upper triangle does work; bi>bj exits immediately
  hn_tile_kernel<<<grid, 256, 0, stream>>>(normed, labels, packed);
  hn_final_kernel<<<1, 256, 0, stream>>>(packed, image_idxs, out);
}